// BinaryTimedPSP_266287973027
// MI455X (gfx1250) — compile-verified
//
#include <hip/hip_runtime.h>
#include <stdint.h>

// ---------------------------------------------------------------------------
// Causal boxcar(100) over time of binary spikes, clamped to [0,1].
// Exact reformulation for binary input: out[t,c] = (t - last_spike[c] < dur).
// Pure HBM-streaming kernel (~512 MB traffic -> ~22 us roofline @ 23.3 TB/s).
// CDNA5 path: GLOBAL_LOAD_ASYNC_TO_LDS_B128 (ASYNCcnt) builds a 16-deep
// per-thread prefetch pipeline into an LDS ring; consumption gated by
// s_wait_asynccnt. Each lane stages/consumes only its own 16 B -> no barriers.
// Inner loop uses purely incremental addressing: issue, wait, ds_load,
// 4 compares, nontemporal b128 store.
// ---------------------------------------------------------------------------

typedef __attribute__((ext_vector_type(4))) float v4f;   // native clang vector

#define T_TOTAL    2048
#define TPB        256                  // threads per block
#define CPB        (TPB * 4)            // columns per block (v4f per thread)
#define NT         8                    // time chunks (grid.y)
#define SLOTS      16                   // LDS ring slots (power of two)
#define SLOT_BYTES (TPB * 16)           // 4096 B per slot
#define RING_BYTES (SLOTS * SLOT_BYTES) // 65536 B
#define DEPTH      15                   // issue-ahead; outstanding = DEPTH+1

__global__ __launch_bounds__(TPB)
void psp_boxcar_kernel(const float* __restrict__ in,
                       const int*   __restrict__ durp,
                       float*       __restrict__ out,
                       int C)
{
    // 16 slots * 256 threads * 16 B = 64 KB LDS ring buffer
    __shared__ v4f buf[SLOTS][TPB];

    const int tid     = threadIdx.x;
    const int colBase = blockIdx.x * CPB + tid * 4;
    const int chunk   = T_TOTAL / NT;            // 256
    const int t0      = blockIdx.y * chunk;
    const int t1      = t0 + chunk;
    const int dur     = *durp;                   // 100

    int tw = t0 - (dur - 1);                     // warm-up start
    if (tw < 0) tw = 0;

    const uint64_t rowBytes = (uint64_t)C * 4u;
    // Low 32 bits of a generic pointer to __shared__ = LDS byte offset.
    const uint32_t ringBase =
        (uint32_t)(uintptr_t)(&buf[0][0]) + (uint32_t)(tid * 16);

    // ---- prologue: issue rows tw .. tw+DEPTH-1 into slots 0..DEPTH-1 -------
    uint64_t gIssue = (uint64_t)(uintptr_t)(in + (uint64_t)tw * (uint64_t)C + colBase);
    uint32_t sIssue = 0;                         // ring byte offset of next issue
    #pragma unroll
    for (int k = 0; k < DEPTH; ++k) {
        asm volatile("global_load_async_to_lds_b128 %0, %1, off"
                     :: "v"(ringBase + sIssue), "v"(gIssue) : "memory");
        gIssue += rowBytes;
        sIssue = (sIssue + SLOT_BYTES) & (RING_BYTES - 1);
    }

    int last0 = -dur, last1 = -dur, last2 = -dur, last3 = -dur;
    int slotIdx = 0;
    const float* pout = out + (uint64_t)tw * (uint64_t)C + colBase;

    // ---- steady state: issue row t+DEPTH, wait oldest, consume row t -------
    for (int t = tw; t < t1; ++t) {
        // Tail (t >= t1-DEPTH) repeatedly issues row t1 (one past the chunk); those writes land only
        // in slots already consumed (slot(t+DEPTH) == slot(t-1)).
        asm volatile("global_load_async_to_lds_b128 %0, %1, off"
                     :: "v"(ringBase + sIssue), "v"(gIssue) : "memory");
        if (t < t1 - DEPTH) gIssue += rowBytes;          // uniform scalar guard
        sIssue = (sIssue + SLOT_BYTES) & (RING_BYTES - 1);

        // outstanding = rows t..t+DEPTH (DEPTH+1); <=DEPTH left => row t done
        asm volatile("s_wait_asynccnt 15" ::: "memory");

        const v4f x = buf[slotIdx][tid];                 // ds_load_b128
        slotIdx = (slotIdx + 1) & (SLOTS - 1);
        if (x.x != 0.0f) last0 = t;
        if (x.y != 0.0f) last1 = t;
        if (x.z != 0.0f) last2 = t;
        if (x.w != 0.0f) last3 = t;

        if (t >= t0) {                                   // uniform warm-up skip
            v4f o;
            o.x = (t - last0) < dur ? 1.0f : 0.0f;
            o.y = (t - last1) < dur ? 1.0f : 0.0f;
            o.z = (t - last2) < dur ? 1.0f : 0.0f;
            o.w = (t - last3) < dur ? 1.0f : 0.0f;
            // write-once stream: non-temporal so it doesn't evict warm-up
            // input rows from L2
            __builtin_nontemporal_store(o, (v4f*)pout);
        }
        pout = (const float*)((const char*)pout + rowBytes);
    }
}

extern "C" void kernel_launch(void* const* d_in, const int* in_sizes, int n_in,
                              void* d_out, int out_size, void* d_ws, size_t ws_size,
                              hipStream_t stream)
{
    const float* in   = (const float*)d_in[0];
    const int*   durp = (const int*)d_in[1];
    float*       out  = (float*)d_out;

    const int total = in_sizes[0];       // T*B*N = 2048*16*2048
    const int C     = total / T_TOTAL;   // 32768 columns

    dim3 grid(C / CPB, NT);              // 32 x 8 = 256 workgroups
    psp_boxcar_kernel<<<grid, TPB, 0, stream>>>(in, durp, out, C);
}